// GausRast_63170378990328
// MI455X (gfx1250) — compile-verified
//
#include <hip/hip_runtime.h>
#include <hip/hip_bf16.h>
#include <math.h>

// ---------------------------------------------------------------------------
// Gaussian splatting rasterizer for MI455X (gfx1250, wave32).
//   K1: per-gaussian preprocess  (conic / opacity / color / depth key / radii)
//   K2: stable depth sort (rank sort, P=512) + visible count
//   K3: tiled blend. 16 pixels per wave; the einsum('phw,pc->chw') is done
//       with V_WMMA_F32_16X16X4_F32 (M=16 pixels, N=3-of-16 channels,
//       K=4 gaussians/step). Transmittance recurrence stays in f32 VALU,
//       split across half-waves and stitched with one __shfl_xor per step.
// ---------------------------------------------------------------------------

#define Hc 256
#define Wc 256
#define FXc 256.0f
#define FYc 256.0f
#define SH_C0f 0.28209479177387814f
#define PMAX 512

// workspace layout (float offsets)
#define OFF_UP   0            // unsorted params  [PMAX*8]  {mx,my,cA,cB,cC,op,0,0}
#define OFF_UC   (PMAX*8)     // unsorted colors  [PMAX*4]
#define OFF_KEY  (PMAX*12)    // depth keys       [PMAX]
#define OFF_SP   (PMAX*13)    // sorted params    [PMAX*8]
#define OFF_SC   (PMAX*21)    // sorted colors    [PMAX*4]
#define OFF_NVIS (PMAX*25)    // 1 int: visible count

typedef __attribute__((ext_vector_type(2))) float v2f;
typedef __attribute__((ext_vector_type(8))) float v8f;

// ------------------------------- kernel 1 ----------------------------------
__global__ void gs_preprocess(const float* __restrict__ points,
                              const float* __restrict__ scales,
                              const float* __restrict__ rots,
                              const float* __restrict__ opacities,
                              const float* __restrict__ shs,
                              const float* __restrict__ vm,
                              float* __restrict__ ws,
                              float* __restrict__ out, int P)
{
    int p = blockIdx.x * blockDim.x + threadIdx.x;
    if (p >= P) return;

    float s0 = expf(scales[p*3+0]);
    float s1 = expf(scales[p*3+1]);
    float s2 = expf(scales[p*3+2]);
    float op = 1.0f / (1.0f + expf(-opacities[p]));

    float qr = rots[p*4+0], qx = rots[p*4+1], qy = rots[p*4+2], qz = rots[p*4+3];
    float inv = rsqrtf(qr*qr + qx*qx + qy*qy + qz*qz);
    qr *= inv; qx *= inv; qy *= inv; qz *= inv;

    float R00 = 1.f - 2.f*(qy*qy + qz*qz), R01 = 2.f*(qx*qy - qr*qz), R02 = 2.f*(qx*qz + qr*qy);
    float R10 = 2.f*(qx*qy + qr*qz), R11 = 1.f - 2.f*(qx*qx + qz*qz), R12 = 2.f*(qy*qz - qr*qx);
    float R20 = 2.f*(qx*qz - qr*qy), R21 = 2.f*(qy*qz + qr*qx), R22 = 1.f - 2.f*(qx*qx + qy*qy);

    float M00 = R00*s0, M01 = R01*s1, M02 = R02*s2;
    float M10 = R10*s0, M11 = R11*s1, M12 = R12*s2;
    float M20 = R20*s0, M21 = R21*s1, M22 = R22*s2;

    float S00 = M00*M00 + M01*M01 + M02*M02;
    float S01 = M00*M10 + M01*M11 + M02*M12;
    float S02 = M00*M20 + M01*M21 + M02*M22;
    float S11 = M10*M10 + M11*M11 + M12*M12;
    float S12 = M10*M20 + M11*M21 + M12*M22;
    float S22 = M20*M20 + M21*M21 + M22*M22;

    float W00 = vm[0], W01 = vm[1], W02 = vm[2],  o0 = vm[3];
    float W10 = vm[4], W11 = vm[5], W12 = vm[6],  o1 = vm[7];
    float W20 = vm[8], W21 = vm[9], W22 = vm[10], o2 = vm[11];

    float px = points[p*3+0], py = points[p*3+1], pz = points[p*3+2];
    float t0 = W00*px + W01*py + W02*pz + o0;
    float t1 = W10*px + W11*py + W12*pz + o1;
    float t2 = W20*px + W21*py + W22*pz + o2;

    bool vis = t2 > 0.2f;
    float tz = vis ? t2 : 1.0f;
    const float lim = 1.3f * 0.5f;
    float txtz = fminf(fmaxf(t0 / tz, -lim), lim) * tz;
    float tytz = fminf(fmaxf(t1 / tz, -lim), lim) * tz;
    float iz = 1.0f / tz, iz2 = iz * iz;

    float J00 = FXc * iz, J02 = -FXc * txtz * iz2;
    float J11 = FYc * iz, J12 = -FYc * tytz * iz2;

    // Tm = J @ Wrot
    float T00 = J00*W00 + J02*W20, T01 = J00*W01 + J02*W21, T02 = J00*W02 + J02*W22;
    float T10 = J11*W10 + J12*W20, T11 = J11*W11 + J12*W21, T12 = J11*W12 + J12*W22;

    // cov = Tm Sigma Tm^T
    float U00 = T00*S00 + T01*S01 + T02*S02;
    float U01 = T00*S01 + T01*S11 + T02*S12;
    float U02 = T00*S02 + T01*S12 + T02*S22;
    float U10 = T10*S00 + T11*S01 + T12*S02;
    float U11 = T10*S01 + T11*S11 + T12*S12;
    float U12 = T10*S02 + T11*S12 + T12*S22;
    float c00 = U00*T00 + U01*T01 + U02*T02;
    float c01 = U00*T10 + U01*T11 + U02*T12;
    float c11 = U10*T10 + U11*T11 + U12*T12;

    float det_orig = c00*c11 - c01*c01;
    float a = c00 + 0.3f, c = c11 + 0.3f, b = c01;
    float det = a*c - b*b;
    vis = vis && (det > 0.0f);
    float det_s = (det > 0.0f) ? det : 1.0f;
    float comp = sqrtf(fmaxf(det_orig / det_s, 0.0f));
    float op_eff = op * comp * (vis ? 1.0f : 0.0f);
    float conA = c / det_s, conB = -b / det_s, conC = a / det_s;
    float mid = 0.5f * (a + c);
    float lam = mid + sqrtf(fmaxf(mid*mid - det, 0.1f));
    float rad = vis ? ceilf(3.0f * sqrtf(lam)) : 0.0f;

    float mx = FXc * t0 * iz + (Wc - 1) * 0.5f;
    float my = FYc * t1 * iz + (Hc - 1) * 0.5f;

    float cr = fmaxf(shs[p*3+0] * SH_C0f + 0.5f, 0.0f);
    float cg = fmaxf(shs[p*3+1] * SH_C0f + 0.5f, 0.0f);
    float cb = fmaxf(shs[p*3+2] * SH_C0f + 0.5f, 0.0f);

    ws[OFF_UP + p*8 + 0] = mx;
    ws[OFF_UP + p*8 + 1] = my;
    ws[OFF_UP + p*8 + 2] = conA;
    ws[OFF_UP + p*8 + 3] = conB;
    ws[OFF_UP + p*8 + 4] = conC;
    ws[OFF_UP + p*8 + 5] = op_eff;
    ws[OFF_UP + p*8 + 6] = 0.0f;
    ws[OFF_UP + p*8 + 7] = 0.0f;
    ws[OFF_UC + p*4 + 0] = cr;
    ws[OFF_UC + p*4 + 1] = cg;
    ws[OFF_UC + p*4 + 2] = cb;
    ws[OFF_UC + p*4 + 3] = 0.0f;
    ws[OFF_KEY + p] = vis ? t2 : __builtin_inff();

    // outputs: viewspace_points (zeros) and radii (as float)
    int vspOff = 3 * Hc * Wc;
    out[vspOff + p*3 + 0] = 0.0f;
    out[vspOff + p*3 + 1] = 0.0f;
    out[vspOff + p*3 + 2] = 0.0f;
    out[vspOff + 3*P + p] = rad;
}

// ------------------------------- kernel 2 ----------------------------------
// Stable ascending rank sort on depth keys (inf for invisible => sorted last),
// scatter params/colors into sorted order, record visible count.
__global__ void gs_sort(float* __restrict__ ws, int P)
{
    __shared__ float keys[PMAX];
    int t = threadIdx.x;
    if (t < P) keys[t] = ws[OFF_KEY + t];
    __syncthreads();
    if (t < P) {
        float k = keys[t];
        int rank = 0;
        for (int q = 0; q < P; ++q) {
            float kq = keys[q];
            rank += ((kq < k) || (kq == k && q < t)) ? 1 : 0;
        }
        #pragma unroll
        for (int j = 0; j < 8; ++j) ws[OFF_SP + rank*8 + j] = ws[OFF_UP + t*8 + j];
        #pragma unroll
        for (int j = 0; j < 4; ++j) ws[OFF_SC + rank*4 + j] = ws[OFF_UC + t*4 + j];
    }
    if (t == 0) {
        const float finf = __builtin_inff();
        int cnt = 0;
        for (int q = 0; q < P; ++q) cnt += (keys[q] < finf) ? 1 : 0;
        ((int*)ws)[OFF_NVIS] = cnt;
    }
}

// ------------------------------- kernel 3 ----------------------------------
// 256 threads = 8 waves; each wave blends 16 consecutive pixels.
// Per step of 4 gaussians:
//   lanes 0-15  compute alpha for gaussians g,g+1 (A-matrix K0,K1)
//   lanes 16-31 compute alpha for gaussians g+2,g+3 (A-matrix K2,K3)
// Transmittance prefix stitched across half-waves with one shfl_xor,
// then one V_WMMA_F32_16X16X4_F32 accumulates wgt[16x4] x col[4x16].
__global__ void __launch_bounds__(256)
gs_raster(const float* __restrict__ ws, float* __restrict__ out)
{
    __shared__ float sP[PMAX * 8];
    __shared__ float sC[PMAX * 4];
    int tid = threadIdx.x;
    for (int i = tid; i < PMAX * 8; i += 256) sP[i] = ws[OFF_SP + i];
    for (int i = tid; i < PMAX * 4; i += 256) sC[i] = ws[OFF_SC + i];
    __syncthreads();

    int nvis   = ((const int*)ws)[OFF_NVIS];
    int gLimit = (nvis + 3) & ~3;

    int lane = tid & 31;
    int wave = tid >> 5;
    int n    = lane & 15;      // WMMA column (channel) / pixel row M
    int hi   = lane >> 4;      // half-wave select

    int pixBase = blockIdx.x * 128 + wave * 16;
    int pix = pixBase + n;
    float fx = (float)(pix & (Wc - 1));
    float fy = (float)(pix >> 8);

    v8f acc = {};              // D/C: 16 pixels x 16 cols (3 channels used)
    float T = 1.0f;            // running transmittance (replicated per half)

    for (int g = 0; g < gLimit; g += 4) {
        int g0 = g + (hi << 1);
        const float* q0 = &sP[g0 * 8];
        const float* q1 = &sP[(g0 + 1) * 8];

        float dx0 = fx - q0[0], dy0 = fy - q0[1];
        float pw0 = -0.5f * (q0[2]*dx0*dx0 + q0[4]*dy0*dy0) - q0[3]*dx0*dy0;
        float al0 = fminf(0.99f, q0[5] * expf(fminf(pw0, 0.0f)));
        float a0  = (pw0 <= 0.0f && al0 >= (1.0f/255.0f)) ? al0 : 0.0f;

        float dx1 = fx - q1[0], dy1 = fy - q1[1];
        float pw1 = -0.5f * (q1[2]*dx1*dx1 + q1[4]*dy1*dy1) - q1[3]*dx1*dy1;
        float al1 = fminf(0.99f, q1[5] * expf(fminf(pw1, 0.0f)));
        float a1  = (pw1 <= 0.0f && al1 >= (1.0f/255.0f)) ? al1 : 0.0f;

        // wave-uniform skip: this 4-gaussian group misses all 16 pixels
        if (__all((a0 == 0.0f) && (a1 == 0.0f))) continue;

        float pr = (1.0f - a0) * (1.0f - a1);   // partial (1-a) product of my half
        float po = __shfl_xor(pr, 16);          // other half's partial product

        float Tb  = hi ? T * po : T;            // exclusive T entering my K-pair
        float w0  = (Tb  >= 1e-4f) ? a0 * Tb  : 0.0f;
        float T1v = Tb * (1.0f - a0);
        float w1  = (T1v >= 1e-4f) ? a1 * T1v : 0.0f;
        T *= pr * po;

        // B 4x16 f32: VGPR0 = rows K0(lanes0-15)/K1(lanes16-31), VGPR1 = K2/K3
        float b0 = (n < 3) ? sC[(g + hi) * 4 + n]     : 0.0f;
        float b1 = (n < 3) ? sC[(g + 2 + hi) * 4 + n] : 0.0f;

        v2f Am = {w0, w1};
        v2f Bm = {b0, b1};
        acc = __builtin_amdgcn_wmma_f32_16x16x4_f32(
            false, Am, false, Bm, (short)0, acc, false, false);

        if (__all(T < 1e-4f)) break;            // saturated: all future wgt == 0
    }

    // D layout: VGPR v, lane -> element (M = v + 8*hi, N = lane&15)
    if (n < 3) {
        int base = pixBase + hi * 8;
        #pragma unroll
        for (int v = 0; v < 8; ++v)
            out[n * (Hc * Wc) + base + v] = acc[v];
    }
}

// ------------------------------- launcher ----------------------------------
extern "C" void kernel_launch(void* const* d_in, const int* in_sizes, int n_in,
                              void* d_out, int out_size, void* d_ws, size_t ws_size,
                              hipStream_t stream)
{
    (void)n_in; (void)out_size; (void)ws_size;
    const float* points    = (const float*)d_in[0];
    const float* scales    = (const float*)d_in[1];
    const float* rots      = (const float*)d_in[2];
    const float* opacities = (const float*)d_in[3];
    const float* shs       = (const float*)d_in[4];
    const float* viewmat   = (const float*)d_in[5];
    float* out = (float*)d_out;
    float* ws  = (float*)d_ws;
    int P = in_sizes[0] / 3;   // 512

    gs_preprocess<<<(P + 255) / 256, 256, 0, stream>>>(
        points, scales, rots, opacities, shs, viewmat, ws, out, P);
    gs_sort<<<1, PMAX, 0, stream>>>(ws, P);
    gs_raster<<<(Hc * Wc) / 128, 256, 0, stream>>>(ws, out);
}